// Preprocess_51024211476486
// MI455X (gfx1250) — compile-verified
//
#include <hip/hip_runtime.h>
#include <math.h>

typedef __attribute__((ext_vector_type(2))) float v2f;
typedef __attribute__((ext_vector_type(8))) float v8f;

// 82 landmarks: lh 468..488, rh 522..542, 40 lip indices; padded to 88 so
// 176 features = 11 groups of 16 (pad entries point at landmark 0 -> safe loads).
__constant__ int LM_TABLE[88] = {
    468,469,470,471,472,473,474,475,476,477,478,479,480,481,482,483,484,485,486,487,488,
    522,523,524,525,526,527,528,529,530,531,532,533,534,535,536,537,538,539,540,541,542,
    61,185,40,39,37,0,267,269,270,409,291,146,91,181,84,17,314,405,321,375,
    78,191,80,81,82,13,312,311,310,415,95,88,178,87,14,317,402,318,324,308,
    0,0,0,0,0,0
};

#define FRAME_STRIDE 1629   // 543 * 3 floats
#define NF 164              // real features (82 landmarks x 2 coords)
#define NGROUP 11           // ceil(164/16)
#define NSTRIP 4            // waves per block, 128 frames each

__global__ __launch_bounds__(128)
void seg_stats_wmma(const float* __restrict__ frames, float* __restrict__ out)
{
    const int seg  = blockIdx.x;      // 0..63
    const int grp  = blockIdx.y;      // 0..10  (16 features each)
    const int lane = threadIdx.x & 31;
    const int wave = threadIdx.x >> 5;   // strip id 0..3

    // Segment frame range: boundaries[i] = floor(i*32767/64)
    const int t0 = (seg * 32767) / 64;
    const int t1 = ((seg + 1) * 32767) / 64;   // exclusive

    // WMMA A-matrix (16x4 f32) lane mapping: lanes 0-15 hold K=0,1 for row m,
    // lanes 16-31 hold K=2,3 for row m.
    const int m   = lane & 15;
    const int kHi = lane >> 4;           // 0 -> frames +0,+1 ; 1 -> frames +2,+3
    const int f   = grp * 16 + m;        // feature id (may be pad >=164)
    const int lm  = LM_TABLE[f >> 1];
    const int co  = f & 1;

    const float* base = frames + (size_t)lm * 3 + co;

    v8f accC = {};   // counts
    v8f accS = {};   // sum x
    v8f accQ = {};   // sum x^2
    const v2f ones = {1.0f, 1.0f};

    // This wave's strip: frames [t0 + 128*wave, t0 + 128*wave + 128)
    int tb = t0 + 128 * wave + 2 * kHi;
    const float* p = base + (size_t)tb * FRAME_STRIDE;

    for (int it = 0; it < 32; ++it) {
        const float xA = p[0];
        const float xB = p[FRAME_STRIDE];
        const bool vA = (tb     < t1) && (xA == xA);
        const bool vB = (tb + 1 < t1) && (xB == xB);
        const float a0 = vA ? xA : 0.0f;
        const float a1 = vB ? xB : 0.0f;

        v2f aV = { vA ? 1.0f : 0.0f, vB ? 1.0f : 0.0f };
        v2f aX = { a0, a1 };
        v2f aQ = { a0 * a0, a1 * a1 };

        // D = A x ones + C : every D column n equals the per-feature K-sum.
        accC = __builtin_amdgcn_wmma_f32_16x16x4_f32(false, aV, false, ones,
                                                     (short)0, accC, false, false);
        accS = __builtin_amdgcn_wmma_f32_16x16x4_f32(false, aX, false, ones,
                                                     (short)0, accS, false, false);
        accQ = __builtin_amdgcn_wmma_f32_16x16x4_f32(false, aQ, false, ones,
                                                     (short)0, accQ, false, false);

        tb += 4;
        p  += 4 * FRAME_STRIDE;
    }

    // D layout (f32 16x16): VGPR e, lanes 0-15 -> M=e ; lanes 16-31 -> M=8+e.
    // All columns identical, so lane 0 has features 0..7, lane 16 has 8..15.
    __shared__ float lds[NSTRIP * 16 * 3];
    if (m == 0) {                       // lanes 0 and 16 only
        const int fl = kHi * 8;
        #pragma unroll
        for (int e = 0; e < 8; ++e) {
            const int idx = (wave * 16 + fl + e) * 3;
            lds[idx + 0] = accC[e];
            lds[idx + 1] = accS[e];
            lds[idx + 2] = accQ[e];
        }
    }
    __syncthreads();

    if (threadIdx.x < 16) {
        const int fl = threadIdx.x;
        float c = 0.0f, s = 0.0f, q = 0.0f;
        #pragma unroll
        for (int st = 0; st < NSTRIP; ++st) {
            const int idx = (st * 16 + fl) * 3;
            c += lds[idx + 0];
            s += lds[idx + 1];
            q += lds[idx + 2];
        }
        const int fg = grp * 16 + fl;
        if (fg < NF) {
            float mean = 0.0f, sd = 0.0f;
            if (c > 0.0f) {
                mean = s / c;
                float var = q / c - mean * mean;
                sd = sqrtf(fmaxf(var, 0.0f));
            }
            if (!(mean == mean)) mean = 0.0f;   // NaN -> 0 (matches reference)
            if (!(sd   == sd))   sd   = 0.0f;
            float* row = out + (size_t)seg * 328;
            row[fg]       = mean;   // stats[:, :82, :]  -> cols [0,164)
            row[164 + fg] = sd;     // stats[:, 82:, :]  -> cols [164,328)
        }
    }
}

extern "C" void kernel_launch(void* const* d_in, const int* in_sizes, int n_in,
                              void* d_out, int out_size, void* d_ws, size_t ws_size,
                              hipStream_t stream)
{
    (void)in_sizes; (void)n_in; (void)out_size; (void)d_ws; (void)ws_size;
    const float* frames = (const float*)d_in[0];
    float* out = (float*)d_out;
    dim3 grid(64, NGROUP);      // 64 segments x 11 feature groups
    seg_stats_wmma<<<grid, 128, 0, stream>>>(frames, out);
}